// NLRidge_16664473108909
// MI455X (gfx1250) — compile-verified
//
#include <hip/hip_runtime.h>
#include <math.h>

// ---------------- problem constants (from reference) ----------------
#define NIMG   2
#define HH     256
#define WW     256
#define PP     7
#define NN     49          // P*P patch vector length
#define HP     250         // H - P + 1
#define GRIDP  64          // reference-grid points per dim (stride 4, aligned corner)
#define NGROUP (GRIDP*GRIDP)   // 4096 groups per image
#define WSR    37          // search window
#define VV     18          // WSR/2
#define NCAND  (WSR*WSR)   // 1369 candidates
#define EPSV   1e-6f
#define M1     18
#define M2     55

typedef __attribute__((ext_vector_type(2))) float v2f;
typedef __attribute__((ext_vector_type(8))) float v8f;

__device__ __forceinline__ int ref_pos(int g) {
    // align_corners(arange(250),4)[::4] -> 0,4,...,248,249
    return (g <= 62) ? 4 * g : (HP - 1);
}

// ---------------- zero / finalize helpers ----------------
__global__ void zero_kernel(float* __restrict__ p, int count) {
    for (int i = blockIdx.x * blockDim.x + threadIdx.x; i < count;
         i += gridDim.x * blockDim.x)
        p[i] = 0.0f;
}

__global__ void finalize_kernel(const float* __restrict__ num,
                                const float* __restrict__ den,
                                float* __restrict__ out, int count) {
    for (int i = blockIdx.x * blockDim.x + threadIdx.x; i < count;
         i += gridDim.x * blockDim.x)
        out[i] = num[i] / den[i];
}

// ---------------- block matching ----------------
// One workgroup (256 threads) per reference patch. 43x43 pixel window staged
// in LDS; 1369 candidate distances in LDS; M rounds of argmin selection
// (ascending distance, tie-break lower candidate index, center = -inf first).
template <int M_>
__global__ void bm_kernel(const float* __restrict__ img, int* __restrict__ idx_out) {
    __shared__ float win[43 * 44];
    __shared__ float dist[NCAND];
    __shared__ float sval[256];
    __shared__ int   sidx[256];

    const int b  = blockIdx.x;          // [img][gk][gl]
    const int n  = b / NGROUP;
    const int g  = b % NGROUP;
    const int gk = g / GRIDP, gl = g % GRIDP;
    const int rr = ref_pos(gk), rc = ref_pos(gl);
    const float* im = img + n * (HH * WW);
    const int tid = threadIdx.x;

    // stage 43x43 search window (origin rr-18, rc-18); OOB pixels unused
    for (int e = tid; e < 43 * 43; e += 256) {
        int r = rr - VV + e / 43, c = rc - VV + e % 43;
        float vpix = 0.0f;
        if (r >= 0 && r < HH && c >= 0 && c < WW) {
            __builtin_prefetch(&im[r * WW + c], 0, 0);
            vpix = im[r * WW + c];
        }
        win[(e / 43) * 44 + (e % 43)] = vpix;
    }
    __syncthreads();

    // 49-dim L2 distance per candidate offset
    for (int cnd = tid; cnd < NCAND; cnd += 256) {
        int dr = cnd / WSR - VV, dc = cnd % WSR - VV;
        int r = rr + dr, c = rc + dc;
        float d;
        if (dr == 0 && dc == 0) {
            d = -INFINITY;                       // self slot: always picked first
        } else if (r < 0 || r > HP - 1 || c < 0 || c > HP - 1) {
            d = INFINITY;                        // inf-padded search area
        } else {
            float s = 0.0f;
            for (int ki = 0; ki < PP; ++ki)
                for (int kj = 0; kj < PP; ++kj) {
                    float diff = win[(VV + dr + ki) * 44 + (VV + dc + kj)]
                               - win[(VV + ki) * 44 + (VV + kj)] + EPSV;
                    s += diff * diff;
                }
            d = sqrtf(s);
        }
        dist[cnd] = d;
    }
    __syncthreads();

    int* outp = idx_out + b * M_;
    for (int m = 0; m < M_; ++m) {
        float bv = INFINITY; int bi = NCAND;
        for (int cnd = tid; cnd < NCAND; cnd += 256) {
            float dv = dist[cnd];
            if (dv < bv || (dv == bv && cnd < bi)) { bv = dv; bi = cnd; }
        }
        sval[tid] = bv; sidx[tid] = bi;
        __syncthreads();
        for (int st = 128; st > 0; st >>= 1) {
            if (tid < st) {
                float ov = sval[tid + st]; int oi = sidx[tid + st];
                if (ov < sval[tid] || (ov == sval[tid] && oi < sidx[tid])) {
                    sval[tid] = ov; sidx[tid] = oi;
                }
            }
            __syncthreads();
        }
        if (tid == 0) {
            int c = sidx[0];
            dist[c] = INFINITY;
            int dr = c / WSR - VV, dc = c % WSR - VV;
            int ir = min(rr + dr, HP - 1), ic = min(rc + dc, HP - 1);
            outp[m] = ir * HP + ic;
        }
        __syncthreads();
    }
}

// ---------------- batched group denoise ----------------
// One workgroup per group. Gram & theta*Y via V_WMMA_F32_16X16X4_F32 tiles
// (f32 accumulate: Cholesky is numerically sensitive), Cholesky + cho_solve
// in LDS with f32 VALU, scatter-aggregation via global f32 atomics.
// M_ = group size (18/55), MP = padded (32/64), NWAVES = waves per block.
template <int M_, int MP, int NWAVES>
__global__ __launch_bounds__(NWAVES * 32)
void denoise_kernel(const float* __restrict__ gsrc,   // source for Gram groups
                    const float* __restrict__ ysrc,   // source for noisy groups
                    const int*   __restrict__ idx,
                    const float* __restrict__ sigma,
                    float* __restrict__ num, float* __restrict__ den,
                    int pass2) {
    constexpr int NTH = NWAVES * 32;
    constexpr int T   = MP / 16;                  // 16x16 tiles per dim

    __shared__ float Xg[MP][64];                  // Gram-source group (reused as theta)
    __shared__ float Yg[MP][64];                  // noisy group
    __shared__ float G [MP][MP];                  // Gram -> Cholesky factor
    __shared__ float wts[MP];
    __shared__ int   pidx[M_];
    float (*Th)[MP] = reinterpret_cast<float (*)[MP]>(&Xg[0][0]);  // MP*MP <= MP*64

    const int b    = blockIdx.x;
    const int n    = b / NGROUP;
    const int tid  = threadIdx.x;
    const int lane = tid & 31, wid = tid >> 5;
    const int lm   = lane & 15, hi = lane >> 4;
    const int* gidx = idx + b * M_;

    // zero-pad everything (WMMA K/M/N padding relies on zeros)
    for (int e = tid; e < MP * 64; e += NTH) { Xg[e / 64][e % 64] = 0.0f; Yg[e / 64][e % 64] = 0.0f; }
    for (int e = tid; e < MP * MP; e += NTH) G[e / MP][e % MP] = 0.0f;
    if (tid < M_) pidx[tid] = gidx[tid];
    __syncthreads();

    // gather both groups (M_ x 49) into LDS
    const float* gim = gsrc + n * (HH * WW);
    const float* yim = ysrc + n * (HH * WW);
    for (int e = tid; e < M_ * NN; e += NTH) {
        int j = e / NN, k = e % NN;
        int pos = pidx[j];
        int off = (pos / HP + k / PP) * WW + (pos % HP + k % PP);
        Xg[j][k] = gim[off];
        Yg[j][k] = yim[off];
    }
    __syncthreads();

    // ---- Gram: G = Xg * Xg^T, K = 49 padded to 52 (13 WMMA k-steps) ----
    for (int t = wid; t < T * T; t += NWAVES) {
        int tm = t / T, tn = t % T;
        v8f acc = {0.f, 0.f, 0.f, 0.f, 0.f, 0.f, 0.f, 0.f};
        for (int k = 0; k < 13; ++k) {
            int kk = 4 * k + 2 * hi;
            v2f a;  a[0]  = Xg[tm * 16 + lm][kk]; a[1]  = Xg[tm * 16 + lm][kk + 1];
            v2f bb; bb[0] = Xg[tn * 16 + lm][kk]; bb[1] = Xg[tn * 16 + lm][kk + 1];
            acc = __builtin_amdgcn_wmma_f32_16x16x4_f32(
                false, a, false, bb, (short)0, acc, false, false);
        }
        for (int r = 0; r < 8; ++r)
            G[tm * 16 + r + 8 * hi][tn * 16 + lm] = acc[r];
    }
    __syncthreads();

    const float nsig2 = (float)NN * sigma[0] * sigma[0];

    // Th <- copy of Gram (overlays Xg; Xg no longer needed)
    for (int e = tid; e < MP * MP; e += NTH) Th[e / MP][e % MP] = 0.0f;
    __syncthreads();
    for (int e = tid; e < M_ * M_; e += NTH) Th[e / M_][e % M_] = G[e / M_][e % M_];
    __syncthreads();
    if (tid < M_) {
        if (pass2) G[tid][tid]  += nsig2;   // chol(XtX + n s^2 I), rhs = XtX
        else       Th[tid][tid] -= nsig2;   // chol(YtY), rhs = YtY - n s^2 I
    }
    __syncthreads();

    // ---- right-looking Cholesky of G (lower) ----
    for (int j = 0; j < M_; ++j) {
        if (tid == 0) G[j][j] = sqrtf(G[j][j]);
        __syncthreads();
        for (int i = j + 1 + tid; i < M_; i += NTH) G[i][j] /= G[j][j];
        __syncthreads();
        int rem = M_ - j - 1;
        for (int e = tid; e < rem * rem; e += NTH) {
            int ii = j + 1 + e / rem, kk = j + 1 + e % rem;
            if (kk <= ii) G[ii][kk] -= G[ii][j] * G[kk][j];
        }
        __syncthreads();
    }

    // ---- cho_solve in place on Th (columns parallel across threads) ----
    for (int c = tid; c < M_; c += NTH) {
        for (int i = 0; i < M_; ++i) {               // L z = rhs
            float s = Th[i][c];
            for (int k2 = 0; k2 < i; ++k2) s -= G[i][k2] * Th[k2][c];
            Th[i][c] = s / G[i][i];
        }
        for (int i = M_ - 1; i >= 0; --i) {          // L^T theta = z
            float s = Th[i][c];
            for (int k2 = i + 1; k2 < M_; ++k2) s -= G[k2][i] * Th[k2][c];
            Th[i][c] = s / G[i][i];
        }
    }
    __syncthreads();

    // ---- weights: 1 / clip(rowsum(theta^2), 1/m, 1) ----
    for (int i = tid; i < M_; i += NTH) {
        float s = 0.0f;
        for (int j = 0; j < M_; ++j) s += Th[i][j] * Th[i][j];
        s = fminf(fmaxf(s, 1.0f / (float)M_), 1.0f);
        wts[i] = 1.0f / s;
    }
    __syncthreads();

    // ---- Xhat = theta * Yg, scatter-aggregate straight from WMMA lanes ----
    constexpr int K2 = (M_ + 3) / 4;                 // K = M_ padded to mult of 4
    float* nb = num + n * (HH * WW);
    float* db = den + n * (HH * WW);
    for (int t = wid; t < T * 4; t += NWAVES) {
        int tm = t / 4, tn = t % 4;
        v8f acc = {0.f, 0.f, 0.f, 0.f, 0.f, 0.f, 0.f, 0.f};
        for (int k = 0; k < K2; ++k) {
            int kk = 4 * k + 2 * hi;
            v2f a;  a[0]  = Th[tm * 16 + lm][kk]; a[1]  = Th[tm * 16 + lm][kk + 1];
            v2f bb; bb[0] = Yg[kk][tn * 16 + lm]; bb[1] = Yg[kk + 1][tn * 16 + lm];
            acc = __builtin_amdgcn_wmma_f32_16x16x4_f32(
                false, a, false, bb, (short)0, acc, false, false);
        }
        for (int r = 0; r < 8; ++r) {
            int Mi = tm * 16 + r + 8 * hi;           // patch index in group
            int Ni = tn * 16 + lm;                   // pixel index in patch
            if (Mi < M_ && Ni < NN) {
                float wv  = wts[Mi];
                int   pos = pidx[Mi];
                int   pr  = pos / HP + Ni / PP;
                int   pc  = pos % HP + Ni % PP;
                atomicAdd(&nb[pr * WW + pc], acc[r] * wv);
                atomicAdd(&db[pr * WW + pc], wv);
            }
        }
    }
}

// ---------------- host-side launch ----------------
extern "C" void kernel_launch(void* const* d_in, const int* in_sizes, int n_in,
                              void* d_out, int out_size, void* d_ws, size_t ws_size,
                              hipStream_t stream) {
    const float* y     = (const float*)d_in[0];   // (2,1,256,256) f32
    const float* sigma = (const float*)d_in[1];   // scalar f32
    float* out = (float*)d_out;                   // (2,1,256,256) f32

    // workspace layout (bytes)
    char* ws = (char*)d_ws;
    int*   idx1 = (int*)  (ws + 0);               // 2*4096*18 ints  = 589824 B
    int*   idx2 = (int*)  (ws + 589824);          // 2*4096*55 ints  = 1802240 B
    float* den1 = (float*)(ws + 2392064);         // 2*65536 f32
    float* num  = (float*)(ws + 2916352);         // 2*65536 f32
    float* den  = (float*)(ws + 3440640);         // 2*65536 f32 (contiguous after num)

    const int NB   = NIMG * NGROUP;               // 8192 groups per pass
    const int NPIX = NIMG * HH * WW;              // 131072

    // -------- pass 1 (m=18, p=7) --------
    zero_kernel<<<512, 256, 0, stream>>>(num, 2 * NPIX);           // num+den
    bm_kernel<M1><<<NB, 256, 0, stream>>>(y, idx1);                // y_mean == y (C=1)
    denoise_kernel<M1, 32, 4><<<NB, 128, 0, stream>>>(y, y, idx1, sigma, num, den, 0);
    finalize_kernel<<<512, 256, 0, stream>>>(num, den, den1, NPIX);

    // -------- pass 2 (m=55, p=7) --------
    zero_kernel<<<512, 256, 0, stream>>>(num, 2 * NPIX);
    bm_kernel<M2><<<NB, 256, 0, stream>>>(den1, idx2);             // x_mean == den1
    denoise_kernel<M2, 64, 8><<<NB, 256, 0, stream>>>(den1, y, idx2, sigma, num, den, 1);
    finalize_kernel<<<512, 256, 0, stream>>>(num, den, out, NPIX);
}